// Quantizer_618475291443
// MI455X (gfx1250) — compile-verified
//
#include <hip/hip_runtime.h>
#include <hip/hip_bf16.h>
#include <math.h>

// ---------------------------------------------------------------------------
// VQ-VAE quantizer for MI455X (gfx1250, wave32, WMMA + async LDS DMA).
// Shapes: K=16384 codes, HID=256, D=256, B=16,H=16,W=16 -> N=4096 vectors.
// Outputs (concat in d_out, float): z_q_out[1048576], loss[1], d[67108864],
//                                   perplexity[1], idx[4096], ploss[1]
// ---------------------------------------------------------------------------

#define KC   16384
#define HID  256
#define DCH  256
#define NV   4096          // B*H*W
#define LDSW 260           // padded LDS row stride (floats) -> conflict-free reads

typedef __attribute__((ext_vector_type(2))) float v2f;
typedef __attribute__((ext_vector_type(8))) float v8f;

static __device__ __forceinline__ v8f wmma_f32(v2f a, v2f b, v8f c) {
  // V_WMMA_F32_16X16X4_F32 : D = A(16x4) * B(4x16) + C(16x16), fp32 throughout
  return __builtin_amdgcn_wmma_f32_16x16x4_f32(
      false, a, false, b, (short)0, c, false, false);
}

// Async DMA one 16-byte chunk global -> LDS (ASYNCcnt-tracked, no VGPR bounce).
// The DS address VGPR gets the low 32 bits of the flat LDS pointer (the shared
// aperture lives in bits 63:32, so addr[31:0] is the in-LDS byte offset).
// Deriving the operand from the real pointer (ptrtoint = address capture)
// combined with the "memory" clobber keeps alias analysis honest, so the
// later ds_load reads of the staged tile are not folded to undef.
static __device__ __forceinline__ void async_g2l_b128(float* ldsDst,
                                                      const float* g) {
  const unsigned int dsOff = (unsigned int)(uintptr_t)ldsDst;
  asm volatile("global_load_async_to_lds_b128 %0, %1, off"
               :: "v"(dsOff), "v"((unsigned long long)(uintptr_t)g)
               : "memory");
}
static __device__ __forceinline__ void wait_async0() {
  asm volatile("s_wait_asynccnt 0x0" ::: "memory");
}

// ---------------- workspace layout (bytes) ----------------
#define WS_CB      0u                      // codebook   [KC*DCH] f32  (16 MB)
#define WS_ZF      16777216u               // zf         [NV*DCH] f32  (4 MB)
#define WS_CNORM   20971520u               // cnorm      [KC]     f32
#define WS_ZNORM   21037056u               // znorm      [NV]     f32
#define WS_MINPK   21053440u               // minpack    [NV]     u64  (8B aligned)
#define WS_ROWSUM  21086208u               // rowsum     [NV]     f32
#define WS_DMIN    21102592u               // dmin       [NV]     f32
#define WS_IDX     21118976u               // idx        [NV]     i32
#define WS_COUNTS  21135360u               // counts     [KC]     f32
#define WS_SCAL    21200896u               // scalars: [0]=mse_sum [1]=plogp [2]=hcnt
#define WS_EMEAN   21200912u               // e_mean     [KC]     f32

// ---------------------------------------------------------------------------
__global__ __launch_bounds__(256)
void k0_init(unsigned long long* minpk, float* counts, float* e_mean, float* scal) {
  int i = blockIdx.x * 256 + threadIdx.x;
  if (i < NV) minpk[i] = 0xFFFFFFFFFFFFFFFFull;
  if (i < KC) { counts[i] = 0.0f; e_mean[i] = 0.0f; }
  if (i < 4)  scal[i] = 0.0f;
}

// ---------------------------------------------------------------------------
// K1: codebook = emb @ hn_w.T + hn_b   (WMMA fp32).  1024 blocks x 256 thr.
// Block async-DMAs 16 rows of emb into LDS; 8 waves x 2 column tiles each.
__global__ __launch_bounds__(256)
void k1_codebook(const float* __restrict__ emb, const float* __restrict__ hnw,
                 const float* __restrict__ hnb, float* __restrict__ cb) {
  __shared__ float lza[16 * LDSW];
  const int rowBase = blockIdx.x * 16;
  const int tid = threadIdx.x;
  #pragma unroll
  for (int c = tid; c < 1024; c += 256) {            // 1024 x b128 chunks
    const int row = c >> 6, col = (c & 63) * 4;
    async_g2l_b128(&lza[row * LDSW + col],
                   emb + (size_t)(rowBase + row) * HID + col);
  }
  wait_async0();
  __syncthreads();
  const int wave = tid >> 5, lane = tid & 31;
  const int half = lane >> 4, l16 = lane & 15;
  for (int ct = 0; ct < 2; ++ct) {
    const int colBase = (wave * 2 + ct) * 16;
    v8f acc = {};
    const float* aptr = &lza[l16 * LDSW];
    const float* bptr = &hnw[(size_t)(colBase + l16) * HID];  // B[k][n]=hn_w[n][k]
    #pragma unroll 8
    for (int dd = 0; dd < HID; dd += 4) {
      v2f a, b;
      a.x = aptr[dd + 2 * half]; a.y = aptr[dd + 2 * half + 1];
      b.x = bptr[dd + 2 * half]; b.y = bptr[dd + 2 * half + 1];
      acc = wmma_f32(a, b, acc);
    }
    const float bias = hnb[colBase + l16];
    #pragma unroll
    for (int r = 0; r < 8; ++r) {
      const int row = rowBase + r + 8 * half;
      cb[(size_t)row * DCH + colBase + l16] = acc[r] + bias;
    }
  }
}

// ---------------------------------------------------------------------------
// K2a: cnorm[k] = sum_d codebook[k,d]^2.   16384 blocks x 256.
__global__ __launch_bounds__(256)
void k2_cnorm(const float* __restrict__ cb, float* __restrict__ cnorm) {
  __shared__ float red[256];
  const int k = blockIdx.x, t = threadIdx.x;
  const float v = cb[(size_t)k * DCH + t];
  red[t] = v * v;
  __syncthreads();
  for (int s = 128; s > 0; s >>= 1) { if (t < s) red[t] += red[t + s]; __syncthreads(); }
  if (t == 0) cnorm[k] = red[0];
}

// K2b: transpose z -> zf[N,D] and znorm[n].   4096 blocks x 256.
__global__ __launch_bounds__(256)
void k2_zf(const float* __restrict__ z, float* __restrict__ zf,
           float* __restrict__ znorm) {
  __shared__ float red[256];
  const int n = blockIdx.x, t = threadIdx.x;
  const int b = n >> 8, h = (n >> 4) & 15, w = n & 15;
  const float v = z[(((size_t)b * DCH + t) * 16 + h) * 16 + w];
  zf[(size_t)n * DCH + t] = v;
  red[t] = v * v;
  __syncthreads();
  for (int s = 128; s > 0; s >>= 1) { if (t < s) red[t] += red[t + s]; __syncthreads(); }
  if (t == 0) znorm[n] = red[0];
}

// ---------------------------------------------------------------------------
// K3: d[n,k] = ||z||^2 + ||c||^2 - 2 z.c   via WMMA fp32, fused argmin.
// Grid = 2048 blocks (256 row-tiles x 8 column slices) x 256 threads.
// Each wave: 16 column tiles of 16x16, 64 WMMA steps each (Kdim=256).
__global__ __launch_bounds__(256)
void k3_dist(const float* __restrict__ zf, const float* __restrict__ cb,
             const float* __restrict__ znorm, const float* __restrict__ cnorm,
             float* __restrict__ out_d, unsigned long long* __restrict__ minpk) {
  __shared__ float lza[16 * LDSW];
  const int rowTile = blockIdx.x >> 3;
  const int slice   = blockIdx.x & 7;
  const int rowBase = rowTile * 16;
  const int tid = threadIdx.x;
  #pragma unroll
  for (int c = tid; c < 1024; c += 256) {            // async DMA A-tile to LDS
    const int row = c >> 6, col = (c & 63) * 4;
    async_g2l_b128(&lza[row * LDSW + col],
                   zf + (size_t)(rowBase + row) * DCH + col);
  }
  wait_async0();
  __syncthreads();
  const int wave = tid >> 5, lane = tid & 31;
  const int half = lane >> 4, l16 = lane & 15;

  float znr[8];
  #pragma unroll
  for (int r = 0; r < 8; ++r) znr[r] = znorm[rowBase + r + 8 * half];

  float minv[8];
  unsigned int mini[8];
  #pragma unroll
  for (int r = 0; r < 8; ++r) { minv[r] = 3.4e38f; mini[r] = 0u; }

  const float* aptr = &lza[l16 * LDSW];
  for (int ct = 0; ct < 16; ++ct) {
    const int colTile = slice * 128 + ct * 8 + wave;    // 0..1023
    const int colBase = colTile * 16;
    v8f acc = {};
    const float* bptr = &cb[(size_t)(colBase + l16) * DCH];  // B[d][k]=cb[k][d]
    #pragma unroll 8
    for (int dd = 0; dd < DCH; dd += 4) {
      v2f a, b;
      a.x = aptr[dd + 2 * half]; a.y = aptr[dd + 2 * half + 1];
      b.x = bptr[dd + 2 * half]; b.y = bptr[dd + 2 * half + 1];
      acc = wmma_f32(a, b, acc);
    }
    const float cn  = cnorm[colBase + l16];
    const unsigned int col = (unsigned int)(colBase + l16);
    #pragma unroll
    for (int r = 0; r < 8; ++r) {
      const float dval = znr[r] + cn - 2.0f * acc[r];
      const int row = rowBase + r + 8 * half;
      out_d[(size_t)row * KC + col] = dval;
      if (dval < minv[r]) { minv[r] = dval; mini[r] = col; }
    }
  }
  // cross-lane min over the 16 lanes holding the same row (xor stays in half)
  #pragma unroll
  for (int r = 0; r < 8; ++r) {
    float v = minv[r];
    unsigned int ii = mini[r];
    #pragma unroll
    for (int m = 8; m >= 1; m >>= 1) {
      float ov = __shfl_xor(v, m, 32);
      unsigned int oi = (unsigned int)__shfl_xor((int)ii, m, 32);
      if (ov < v || (ov == v && oi < ii)) { v = ov; ii = oi; }
    }
    if (l16 == 0) {
      unsigned int u = __float_as_uint(v);
      u = (u & 0x80000000u) ? ~u : (u | 0x80000000u);  // order-preserving map
      unsigned long long key = ((unsigned long long)u << 32) | (unsigned long long)ii;
      atomicMin(&minpk[rowBase + r + 8 * half], key);
    }
  }
}

// ---------------------------------------------------------------------------
// K4: per-row: decode (dmin, idx); rowsum = sum_k exp(-(d-dmin)/0.1);
//     emit idx (float) and histogram counts.   4096 blocks x 256.
__global__ __launch_bounds__(256)
void k4_row(const float* __restrict__ d, const unsigned long long* __restrict__ minpk,
            float* __restrict__ rowsum, float* __restrict__ dminf,
            int* __restrict__ idx_i, float* __restrict__ out_idx,
            float* __restrict__ counts) {
  __shared__ float red[256];
  __shared__ float s_dmin;
  __shared__ int   s_idx;
  const int n = blockIdx.x, t = threadIdx.x;
  if (t == 0) {
    const unsigned long long key = minpk[n];
    unsigned int hi = (unsigned int)(key >> 32);
    hi = (hi & 0x80000000u) ? (hi ^ 0x80000000u) : ~hi;   // inverse order map
    s_dmin = __uint_as_float(hi);
    s_idx  = (int)(key & 0xFFFFFFFFull);
  }
  __syncthreads();
  const float dmin = s_dmin;
  float s = 0.0f;
  for (int k = t; k < KC; k += 256)
    s += expf((dmin - d[(size_t)n * KC + k]) * 10.0f);
  red[t] = s;
  __syncthreads();
  for (int st = 128; st > 0; st >>= 1) { if (t < st) red[t] += red[t + st]; __syncthreads(); }
  if (t == 0) {
    rowsum[n]  = red[0];
    dminf[n]   = dmin;
    idx_i[n]   = s_idx;
    out_idx[n] = (float)s_idx;
    atomicAdd(&counts[s_idx], 1.0f);
  }
}

// ---------------------------------------------------------------------------
// K5: partial e_mean[k] += sum over an N-slice of softmax(-d/0.1)[n,k].
// 1024 blocks (64 k-blocks x 16 n-slices) x 256 threads; coalesced d reads.
__global__ __launch_bounds__(256)
void k5_col(const float* __restrict__ d, const float* __restrict__ dminf,
            const float* __restrict__ rowsum, float* __restrict__ e_mean) {
  const int t  = threadIdx.x;
  const int kb = blockIdx.x & 63;
  const int ns = blockIdx.x >> 6;
  const int k  = kb * 256 + t;
  float e = 0.0f;
  const int n0 = ns * 256;
  for (int n = n0; n < n0 + 256; ++n)
    e += expf((dminf[n] - d[(size_t)n * KC + k]) * 10.0f) / rowsum[n];
  atomicAdd(&e_mean[k], e);
}

// K5b: plogp = sum_k e*log(e+eps), e = e_mean[k]/N.   64 blocks x 256.
__global__ __launch_bounds__(256)
void k5b_plogp(const float* __restrict__ e_mean, float* __restrict__ scal) {
  __shared__ float red[256];
  const int t = threadIdx.x;
  const int k = blockIdx.x * 256 + t;
  const float e = e_mean[k] * (1.0f / (float)NV);
  red[t] = e * logf(e + 1e-10f);
  __syncthreads();
  for (int s = 128; s > 0; s >>= 1) { if (t < s) red[t] += red[t + s]; __syncthreads(); }
  if (t == 0) atomicAdd(&scal[1], red[0]);
}

// K6: histogram entropy for non-differentiable perplexity. 64 blocks x 256.
__global__ __launch_bounds__(256)
void k6_hist(const float* __restrict__ counts, float* __restrict__ scal) {
  __shared__ float red[256];
  const int t = threadIdx.x;
  const int k = blockIdx.x * 256 + t;
  const float p = counts[k] * (1.0f / (float)NV);
  red[t] = p * logf(p + 1e-10f);
  __syncthreads();
  for (int s = 128; s > 0; s >>= 1) { if (t < s) red[t] += red[t + s]; __syncthreads(); }
  if (t == 0) atomicAdd(&scal[2], red[0]);
}

// K7: z_q gather (straight-through value = codebook row) + MSE accumulation.
// 4096 blocks x 256.
__global__ __launch_bounds__(256)
void k7_zq(const float* __restrict__ cb, const float* __restrict__ zf,
           const int* __restrict__ idx_i, float* __restrict__ out_zq,
           float* __restrict__ scal) {
  __shared__ float red[256];
  const int n = blockIdx.x, t = threadIdx.x;
  const int b = n >> 8, h = (n >> 4) & 15, w = n & 15;
  const float c  = cb[(size_t)idx_i[n] * DCH + t];
  const float zp = zf[(size_t)n * DCH + t];
  out_zq[(((size_t)b * DCH + t) * 16 + h) * 16 + w] = c;  // z_q value (ST grad only)
  const float diff = c - zp;
  red[t] = diff * diff;
  __syncthreads();
  for (int s = 128; s > 0; s >>= 1) { if (t < s) red[t] += red[t + s]; __syncthreads(); }
  if (t == 0) atomicAdd(&scal[0], red[0]);
}

// K8: final scalars.
__global__ void k8_final(const float* __restrict__ scal, float* __restrict__ out_loss,
                         float* __restrict__ out_perp, float* __restrict__ out_ploss) {
  const float mse   = scal[0] * (1.0f / (float)(NV * DCH));
  const float ploss = expf(scal[1]);
  *out_loss  = 1.25f * mse + 0.1f * ploss;   // ALPHA*mse + BETA*mse + GAMMA*ploss
  *out_perp  = expf(-scal[2]);
  *out_ploss = ploss;
}

// ---------------------------------------------------------------------------
extern "C" void kernel_launch(void* const* d_in, const int* in_sizes, int n_in,
                              void* d_out, int out_size, void* d_ws, size_t ws_size,
                              hipStream_t stream) {
  const float* z   = (const float*)d_in[0];   // [16,256,16,16]
  const float* emb = (const float*)d_in[1];   // [16384,256]
  const float* hnw = (const float*)d_in[2];   // [256,256]
  const float* hnb = (const float*)d_in[3];   // [256]

  char* ws = (char*)d_ws;
  float*              cb     = (float*)(ws + WS_CB);
  float*              zf     = (float*)(ws + WS_ZF);
  float*              cnorm  = (float*)(ws + WS_CNORM);
  float*              znorm  = (float*)(ws + WS_ZNORM);
  unsigned long long* minpk  = (unsigned long long*)(ws + WS_MINPK);
  float*              rowsum = (float*)(ws + WS_ROWSUM);
  float*              dminf  = (float*)(ws + WS_DMIN);
  int*                idx_i  = (int*)(ws + WS_IDX);
  float*              counts = (float*)(ws + WS_COUNTS);
  float*              scal   = (float*)(ws + WS_SCAL);
  float*              e_mean = (float*)(ws + WS_EMEAN);

  float* out = (float*)d_out;
  float* out_zq    = out;                       // 1,048,576
  float* out_loss  = out + 1048576;             // 1
  float* out_d     = out + 1048577;             // 67,108,864
  float* out_perp  = out_d + (size_t)NV * KC;   // 1
  float* out_idx   = out_perp + 1;              // 4096
  float* out_ploss = out_idx + NV;              // 1

  k0_init    <<<64,   256, 0, stream>>>(minpk, counts, e_mean, scal);
  k1_codebook<<<1024, 256, 0, stream>>>(emb, hnw, hnb, cb);
  k2_cnorm   <<<KC,   256, 0, stream>>>(cb, cnorm);
  k2_zf      <<<NV,   256, 0, stream>>>(z, zf, znorm);
  k3_dist    <<<2048, 256, 0, stream>>>(zf, cb, znorm, cnorm, out_d, minpk);
  k4_row     <<<NV,   256, 0, stream>>>(out_d, minpk, rowsum, dminf, idx_i, out_idx, counts);
  k5_col     <<<1024, 256, 0, stream>>>(out_d, dminf, rowsum, e_mean);
  k5b_plogp  <<<64,   256, 0, stream>>>(e_mean, scal);
  k6_hist    <<<64,   256, 0, stream>>>(counts, scal);
  k7_zq      <<<NV,   256, 0, stream>>>(cb, zf, idx_i, out_zq, scal);
  k8_final   <<<1,    1,   0, stream>>>(scal, out_loss, out_perp, out_ploss);
}